// SEGNNLayer_19937238188636
// MI455X (gfx1250) — compile-verified
//
#include <hip/hip_runtime.h>

// SEGNN layer for MI455X (gfx1250): fp32 WMMA 16x16x4 path.
// Edge kernel: 16 edges/wave, gather -> TP -> [16x128]@[128x64] (+3 vec GEMMs)
//   -> gate -> [16x64]@[64x64] (+3 vec GEMMs) -> gate -> atomic scatter-add.
// Node kernel: same structure + plain TP-linear + residual.

typedef __attribute__((ext_vector_type(2))) float v2f;
typedef __attribute__((ext_vector_type(8))) float v8f;

#define RSQRT3   0.57735026918962576f
#define INV_K128 0.08838834764831845f   // 1/sqrt(128)
#define INV_K64  0.125f                 // 1/sqrt(64)

__device__ __forceinline__ v8f wmma4(v2f a, v2f b, v8f c) {
  // D(16x16,f32) = A(16x4,f32) x B(4x16,f32) + C
  return __builtin_amdgcn_wmma_f32_16x16x4_f32(
      /*neg_a=*/false, a, /*neg_b=*/false, b,
      /*c_mod=*/(short)0, c, /*reuse_a=*/false, /*reuse_b=*/false);
}

__device__ __forceinline__ v8f vzero8() {
  v8f z;
#pragma unroll
  for (int j = 0; j < 8; ++j) z[j] = 0.0f;
  return z;
}

__device__ __forceinline__ float sigmoidf_(float x) {
  return 1.0f / (1.0f + __expf(-x));
}

// GEMM over the implicit tensor-product matrices:
//   ts[16 x (CS+CV)] = [ fs*a0 | dot(fv, a1)/sqrt3 ]   (scalar A)
//   tv_i[16 x (CS+CV)] = [ fs*a1_i | fv_i*a0 ]          (vector A, comp i)
// times LDS weights ws[(CS+CV) x nouts], wv[(CS+CV) x noutv].
// fs: LDS [16][CS]; fv: LDS [16][CV][3]; a0: LDS [16]; a1: LDS [16][3].
template <int CS, int CV, int NT_S, int NT_V>
__device__ __forceinline__ void tp_gemm(const float* fs, const float* fv,
                                        const float* a0v, const float* a1v,
                                        const float* ws, const int nouts,
                                        const float* wv, const int noutv,
                                        v8f (&acc_s)[NT_S], v8f (&acc_v)[3][NT_V],
                                        const int lane) {
  const int r  = lane & 15;   // row (edge/node in tile) for A; col for B
  const int kp = lane >> 4;   // k-pair selector (K 0,1 vs 2,3 inside a 4-step)
  const float az  = a0v[r];
  const float a1x = a1v[r * 3 + 0];
  const float a1y = a1v[r * 3 + 1];
  const float a1z = a1v[r * 3 + 2];
  constexpr int K = CS + CV;
#pragma unroll 4
  for (int kk = 0; kk < K / 4; ++kk) {
    const int k0 = kk * 4 + kp * 2;
    v2f a_ts, a_t0, a_t1, a_t2;
    if (kk * 4 < CS) {  // uniform across wave: CS % 4 == 0
#pragma unroll
      for (int t = 0; t < 2; ++t) {
        const float x  = fs[r * CS + k0 + t];
        const float ts = x * az, t0 = x * a1x, t1 = x * a1y, t2 = x * a1z;
        if (t == 0) { a_ts.x = ts; a_t0.x = t0; a_t1.x = t1; a_t2.x = t2; }
        else        { a_ts.y = ts; a_t0.y = t0; a_t1.y = t1; a_t2.y = t2; }
      }
    } else {
#pragma unroll
      for (int t = 0; t < 2; ++t) {
        const int   c  = k0 + t - CS;
        const float vx = fv[(r * CV + c) * 3 + 0];
        const float vy = fv[(r * CV + c) * 3 + 1];
        const float vz = fv[(r * CV + c) * 3 + 2];
        const float ts = (vx * a1x + vy * a1y + vz * a1z) * RSQRT3;
        const float t0 = vx * az, t1 = vy * az, t2 = vz * az;
        if (t == 0) { a_ts.x = ts; a_t0.x = t0; a_t1.x = t1; a_t2.x = t2; }
        else        { a_ts.y = ts; a_t0.y = t0; a_t1.y = t1; a_t2.y = t2; }
      }
    }
#pragma unroll
    for (int nt = 0; nt < NT_S; ++nt) {
      v2f b;
      b.x = ws[(k0 + 0) * nouts + nt * 16 + r];
      b.y = ws[(k0 + 1) * nouts + nt * 16 + r];
      acc_s[nt] = wmma4(a_ts, b, acc_s[nt]);
    }
#pragma unroll
    for (int nt = 0; nt < NT_V; ++nt) {
      v2f b;
      b.x = wv[(k0 + 0) * noutv + nt * 16 + r];
      b.y = wv[(k0 + 1) * noutv + nt * 16 + r];
      acc_v[0][nt] = wmma4(a_t0, b, acc_v[0][nt]);
      acc_v[1][nt] = wmma4(a_t1, b, acc_v[1][nt]);
      acc_v[2][nt] = wmma4(a_t2, b, acc_v[2][nt]);
    }
  }
}

// e3nn gate: out_s = silu(s[:,:cout]); v *= sigmoid(s[:,cout:]).
// Gate tile layout matches vector tile layout exactly (same lane, same VGPR),
// so this is pure elementwise VALU. Results stored to per-wave LDS for the
// next GEMM's A-fragment construction.
template <int NT_V>
__device__ __forceinline__ void gate_store_lds(const v8f (&acc_s)[2 * NT_V],
                                               const v8f (&acc_v)[3][NT_V],
                                               const float* bs, const float inv,
                                               const int lane, float* ms, float* mv) {
  const int r = lane & 15, kp = lane >> 4;
  constexpr int cout = NT_V * 16;
#pragma unroll
  for (int nt = 0; nt < NT_V; ++nt) {
#pragma unroll
    for (int j = 0; j < 8; ++j) {
      const int row = j + 8 * kp;
      const int c   = nt * 16 + r;
      const float s = acc_s[nt][j] * inv + bs[c];
      const float g = acc_s[NT_V + nt][j] * inv + bs[cout + c];
      const float sil  = s * sigmoidf_(s);
      const float gate = sigmoidf_(g);
      ms[row * cout + c] = sil;
#pragma unroll
      for (int i = 0; i < 3; ++i)
        mv[(row * cout + c) * 3 + i] = acc_v[i][nt][j] * inv * gate;
    }
  }
}

__global__ void segnn_zero_kernel(float* __restrict__ p, const int n) {
  const int i = blockIdx.x * blockDim.x + threadIdx.x;
  if (i < n) p[i] = 0.0f;
}

// ---------------------------------------------------------------------------
// Edge (message) kernel: one wave per 16-edge tile, 4 waves per block.
// ---------------------------------------------------------------------------
__global__ __launch_bounds__(128) void segnn_edge_kernel(
    const float* __restrict__ node_s, const float* __restrict__ node_v,
    const float* __restrict__ edge_attr_s, const float* __restrict__ edge_attr_v,
    const float* __restrict__ m0_ws, const float* __restrict__ m0_bs,
    const float* __restrict__ m0_wv, const float* __restrict__ m1_ws,
    const float* __restrict__ m1_bs, const float* __restrict__ m1_wv,
    const int* __restrict__ senders, const int* __restrict__ receivers,
    float* __restrict__ ag_s, float* __restrict__ ag_v, const int E) {
  extern __shared__ float smem[];
  // Block-shared weights (18560 floats)
  float* w0s = smem;               // m0_ws [128][64]
  float* w0v = w0s + 8192;         // m0_wv [128][32]
  float* b0  = w0v + 4096;         // m0_bs [64]
  float* w1s = b0 + 64;            // m1_ws [64][64]
  float* w1v = w1s + 4096;         // m1_wv [64][32]
  float* b1  = w1v + 2048;         // m1_bs [64]
  float* wbase = b1 + 64;
  const int wid  = threadIdx.x >> 5;
  const int lane = threadIdx.x & 31;
  // Per-wave scratch (4192 floats each)
  float* fs = wbase + wid * 4192;  // xs [16][64]   (later ms [16][32])
  float* fv = fs + 1024;           // xv [16][64][3] (later mv [16][32][3])
  float* a0 = fv + 3072;           // edge_attr_s [16]
  float* a1 = a0 + 16;             // edge_attr_v [16][3]
  int*   snd = (int*)(a1 + 48);    // [16]
  int*   rcv = snd + 16;           // [16]

  for (int i = threadIdx.x; i < 8192; i += 128) w0s[i] = m0_ws[i];
  for (int i = threadIdx.x; i < 4096; i += 128) w0v[i] = m0_wv[i];
  for (int i = threadIdx.x; i < 4096; i += 128) w1s[i] = m1_ws[i];
  for (int i = threadIdx.x; i < 2048; i += 128) w1v[i] = m1_wv[i];
  if (threadIdx.x < 64)       b0[threadIdx.x]      = m0_bs[threadIdx.x];
  else if (threadIdx.x < 128) b1[threadIdx.x - 64] = m1_bs[threadIdx.x - 64];

  // --- gather edge tile into per-wave LDS (wave-internal DS ordering) ---
  const int ebase = (blockIdx.x * 4 + wid) * 16;
  if (lane < 16) {
    const int e = min(ebase + lane, E - 1);
    snd[lane] = senders[e];
    rcv[lane] = receivers[e];
    a0[lane]  = edge_attr_s[e];
    a1[lane * 3 + 0] = edge_attr_v[e * 3 + 0];
    a1[lane * 3 + 1] = edge_attr_v[e * 3 + 1];
    a1[lane * 3 + 2] = edge_attr_v[e * 3 + 2];
  }
  for (int idx = lane; idx < 16 * 64; idx += 32) {  // xs = [s_s | s_r]
    const int row = idx >> 6, c = idx & 63;
    const int n = (c < 32) ? snd[row] : rcv[row];
    fs[idx] = node_s[n * 32 + (c & 31)];
  }
  for (int idx = lane; idx < 16 * 192; idx += 32) { // xv = [v_s | v_r], [row][c][i]
    const int row = idx / 192;
    const int t   = idx - row * 192;
    const int c   = t / 3, i = t - c * 3;
    const int n = (c < 32) ? snd[row] : rcv[row];
    fv[idx] = node_v[(n * 32 + (c & 31)) * 3 + i];
  }
  __syncthreads();  // weights visible to all waves

  // --- block m0: K=128 -> s[16x64], v[16x32x3]; gate -> ms/mv in LDS ---
  v8f acc_s[4], acc_v[3][2];
#pragma unroll
  for (int nt = 0; nt < 4; ++nt) acc_s[nt] = vzero8();
#pragma unroll
  for (int i = 0; i < 3; ++i)
#pragma unroll
    for (int nt = 0; nt < 2; ++nt) acc_v[i][nt] = vzero8();
  tp_gemm<64, 64, 4, 2>(fs, fv, a0, a1, w0s, 64, w0v, 32, acc_s, acc_v, lane);
  gate_store_lds<2>(acc_s, acc_v, b0, INV_K128, lane, fs, fv);

  // --- block m1: K=64 -> s[16x64], v[16x32x3]; gate + scatter-add ---
  v8f acc2_s[4], acc2_v[3][2];
#pragma unroll
  for (int nt = 0; nt < 4; ++nt) acc2_s[nt] = vzero8();
#pragma unroll
  for (int i = 0; i < 3; ++i)
#pragma unroll
    for (int nt = 0; nt < 2; ++nt) acc2_v[i][nt] = vzero8();
  tp_gemm<32, 32, 4, 2>(fs, fv, a0, a1, w1s, 64, w1v, 32, acc2_s, acc2_v, lane);

  const int r = lane & 15, kp = lane >> 4;
#pragma unroll
  for (int nt = 0; nt < 2; ++nt) {
#pragma unroll
    for (int j = 0; j < 8; ++j) {
      const int row = j + 8 * kp;
      if (ebase + row < E) {
        const int c = nt * 16 + r;
        const float s = acc2_s[nt][j] * INV_K64 + b1[c];
        const float g = acc2_s[2 + nt][j] * INV_K64 + b1[32 + c];
        const float sil  = s * sigmoidf_(s);
        const float gate = sigmoidf_(g);
        const int n = rcv[row];
        atomicAdd(&ag_s[n * 32 + c], sil);
#pragma unroll
        for (int i = 0; i < 3; ++i)
          atomicAdd(&ag_v[(n * 32 + c) * 3 + i],
                    acc2_v[i][nt][j] * INV_K64 * gate);
      }
    }
  }
}

// ---------------------------------------------------------------------------
// Node (update) kernel: one wave per 16-node tile, 4 waves per block.
// ---------------------------------------------------------------------------
__global__ __launch_bounds__(128) void segnn_node_kernel(
    const float* __restrict__ node_s, const float* __restrict__ node_v,
    const float* __restrict__ node_attr_s, const float* __restrict__ node_attr_v,
    const float* __restrict__ u0_ws, const float* __restrict__ u0_bs,
    const float* __restrict__ u0_wv, const float* __restrict__ u1_ws,
    const float* __restrict__ u1_bs, const float* __restrict__ u1_wv,
    const float* __restrict__ ag_s, const float* __restrict__ ag_v,
    float* __restrict__ out, const int N) {
  extern __shared__ float smem[];
  float* w0s = smem;              // u0_ws [128][64]
  float* w0v = w0s + 8192;        // u0_wv [128][32]
  float* b0  = w0v + 4096;        // u0_bs [64]
  float* w1s = b0 + 64;           // u1_ws [64][32]
  float* w1v = w1s + 2048;        // u1_wv [64][32]
  float* b1  = w1v + 2048;        // u1_bs [32]
  float* wbase = b1 + 32;         // 16480
  const int wid  = threadIdx.x >> 5;
  const int lane = threadIdx.x & 31;
  float* fs = wbase + wid * 4192; // us [16][64] (later [16][32])
  float* fv = fs + 1024;          // uv [16][64][3] (later [16][32][3])
  float* a0 = fv + 3072;          // node_attr_s [16]
  float* a1 = a0 + 16;            // node_attr_v [16][3]

  for (int i = threadIdx.x; i < 8192; i += 128) w0s[i] = u0_ws[i];
  for (int i = threadIdx.x; i < 4096; i += 128) w0v[i] = u0_wv[i];
  for (int i = threadIdx.x; i < 2048; i += 128) w1s[i] = u1_ws[i];
  for (int i = threadIdx.x; i < 2048; i += 128) w1v[i] = u1_wv[i];
  if (threadIdx.x < 64)      b0[threadIdx.x]      = u0_bs[threadIdx.x];
  else if (threadIdx.x < 96) b1[threadIdx.x - 64] = u1_bs[threadIdx.x - 64];

  const int nbase = (blockIdx.x * 4 + wid) * 16;
  if (lane < 16) {
    const int n = min(nbase + lane, N - 1);
    a0[lane] = node_attr_s[n];
    a1[lane * 3 + 0] = node_attr_v[n * 3 + 0];
    a1[lane * 3 + 1] = node_attr_v[n * 3 + 1];
    a1[lane * 3 + 2] = node_attr_v[n * 3 + 2];
  }
  for (int idx = lane; idx < 16 * 64; idx += 32) {  // us = [node_s | ag_s]
    const int row = idx >> 6, c = idx & 63;
    const int n = min(nbase + row, N - 1);
    fs[idx] = (c < 32) ? node_s[n * 32 + c] : ag_s[n * 32 + (c - 32)];
  }
  for (int idx = lane; idx < 16 * 192; idx += 32) { // uv = [node_v | ag_v]
    const int row = idx / 192;
    const int t   = idx - row * 192;
    const int c   = t / 3, i = t - c * 3;
    const int n = min(nbase + row, N - 1);
    fv[idx] = (c < 32) ? node_v[(n * 32 + c) * 3 + i]
                       : ag_v[(n * 32 + (c - 32)) * 3 + i];
  }
  __syncthreads();

  // --- u0 gated block: K=128 ---
  v8f acc_s[4], acc_v[3][2];
#pragma unroll
  for (int nt = 0; nt < 4; ++nt) acc_s[nt] = vzero8();
#pragma unroll
  for (int i = 0; i < 3; ++i)
#pragma unroll
    for (int nt = 0; nt < 2; ++nt) acc_v[i][nt] = vzero8();
  tp_gemm<64, 64, 4, 2>(fs, fv, a0, a1, w0s, 64, w0v, 32, acc_s, acc_v, lane);
  gate_store_lds<2>(acc_s, acc_v, b0, INV_K128, lane, fs, fv);

  // --- u1 plain TP-linear: K=64 -> s[16x32], v[16x32x3]; residual + store ---
  v8f acc3_s[2], acc3_v[3][2];
#pragma unroll
  for (int nt = 0; nt < 2; ++nt) acc3_s[nt] = vzero8();
#pragma unroll
  for (int i = 0; i < 3; ++i)
#pragma unroll
    for (int nt = 0; nt < 2; ++nt) acc3_v[i][nt] = vzero8();
  tp_gemm<32, 32, 2, 2>(fs, fv, a0, a1, w1s, 32, w1v, 32, acc3_s, acc3_v, lane);

  const int r = lane & 15, kp = lane >> 4;
#pragma unroll
  for (int nt = 0; nt < 2; ++nt) {
#pragma unroll
    for (int j = 0; j < 8; ++j) {
      const int row = j + 8 * kp;
      const int n   = nbase + row;
      if (n < N) {
        const int c = nt * 16 + r;
        const float s = acc3_s[nt][j] * INV_K64 + b1[c];
        out[n * 128 + c] = node_s[n * 32 + c] + s;   // out_s
#pragma unroll
        for (int i = 0; i < 3; ++i)                  // out_v, [NV,3] flattened
          out[n * 128 + 32 + c * 3 + i] =
              node_v[(n * 32 + c) * 3 + i] + acc3_v[i][nt][j] * INV_K64;
      }
    }
  }
}

extern "C" void kernel_launch(void* const* d_in, const int* in_sizes, int n_in,
                              void* d_out, int out_size, void* d_ws, size_t ws_size,
                              hipStream_t stream) {
  (void)n_in; (void)out_size; (void)ws_size;
  const float* node_s      = (const float*)d_in[0];
  const float* node_v      = (const float*)d_in[1];
  const float* node_attr_s = (const float*)d_in[2];
  const float* node_attr_v = (const float*)d_in[3];
  const float* edge_attr_s = (const float*)d_in[4];
  const float* edge_attr_v = (const float*)d_in[5];
  const float* m0_ws = (const float*)d_in[6];
  const float* m0_bs = (const float*)d_in[7];
  const float* m0_wv = (const float*)d_in[8];
  const float* m1_ws = (const float*)d_in[9];
  const float* m1_bs = (const float*)d_in[10];
  const float* m1_wv = (const float*)d_in[11];
  const float* u0_ws = (const float*)d_in[12];
  const float* u0_bs = (const float*)d_in[13];
  const float* u0_wv = (const float*)d_in[14];
  const float* u1_ws = (const float*)d_in[15];
  const float* u1_bs = (const float*)d_in[16];
  const float* u1_wv = (const float*)d_in[17];
  const int* senders   = (const int*)d_in[18];
  const int* receivers = (const int*)d_in[19];

  const int E = in_sizes[18];
  const int N = in_sizes[0] / 32;  // node_s is [N, 32]

  float* ag_s = (float*)d_ws;                 // [N,32]
  float* ag_v = ag_s + (size_t)N * 32;        // [N,32,3]
  float* out  = (float*)d_out;                // [N,128]

  // Zero aggregation workspace (deterministic each call).
  const int ztot = N * 128;
  segnn_zero_kernel<<<(ztot + 255) / 256, 256, 0, stream>>>(ag_s, ztot);

  // Edge kernel: 16 edges/wave * 4 waves/block.
  const size_t lds_edge = (size_t)(18560 + 4 * 4192) * sizeof(float);
  segnn_edge_kernel<<<(E + 63) / 64, 128, lds_edge, stream>>>(
      node_s, node_v, edge_attr_s, edge_attr_v,
      m0_ws, m0_bs, m0_wv, m1_ws, m1_bs, m1_wv,
      senders, receivers, ag_s, ag_v, E);

  // Node kernel: 16 nodes/wave * 4 waves/block.
  const size_t lds_node = (size_t)(16480 + 4 * 4192) * sizeof(float);
  segnn_node_kernel<<<(N + 63) / 64, 128, lds_node, stream>>>(
      node_s, node_v, node_attr_s, node_attr_v,
      u0_ws, u0_bs, u0_wv, u1_ws, u1_bs, u1_wv,
      ag_s, ag_v, out, N);
}